// RasterPoints_19421842112756
// MI455X (gfx1250) — compile-verified
//
#include <hip/hip_runtime.h>
#include <hip/hip_bf16.h>

// RasterPoints: B=8, T=50, N=50 (P=25), H=W=100.
// out[b,t,h,w,p] = 1.0 where (h,w) == (trunc(y/res_y+org0), trunc(x/res_x+org1)), else 0.
// Output is 400 MB -> memset-bound (~17us floor at 23.3 TB/s). Fill with 128-bit
// non-temporal stores (4 per loop iter via immediate offsets), then tiny scatter.

typedef __attribute__((ext_vector_type(4))) float float4v;

// Each block-iteration zeroes one contiguous chunk of 1024 float4s (16 KB).
// Thread t stores at chunkBase + t and three more at +256/+512/+768 float4s
// (= +4096/+8192/+12288 bytes, compile-time immediate offsets -> 4 clauseable
// global_store_b128 th:TH_STORE_NT off a single address register).
__global__ void raster_fill_zero(float4v* __restrict__ out, long long nchunks) {
    float4v z = {0.0f, 0.0f, 0.0f, 0.0f};
    long long chunk = blockIdx.x;
    long long stride = gridDim.x;
    for (; chunk < nchunks; chunk += stride) {
        float4v* base = out + chunk * 1024 + threadIdx.x;
        __builtin_nontemporal_store(z, base);
        __builtin_nontemporal_store(z, base + 256);
        __builtin_nontemporal_store(z, base + 512);
        __builtin_nontemporal_store(z, base + 768);
    }
}

__global__ void raster_fill_zero_tail(float* __restrict__ out, long long start, long long n) {
    long long i = start + (long long)blockIdx.x * blockDim.x + threadIdx.x;
    if (i < n) __builtin_nontemporal_store(0.0f, &out[i]);
}

__global__ void raster_scatter(const float* __restrict__ x,
                               const float* __restrict__ resolution,
                               const float* __restrict__ origin,
                               float* __restrict__ out,
                               int BT, int P, int H, int W) {
    int idx = blockIdx.x * blockDim.x + threadIdx.x;   // over BT*P
    if (idx >= BT * P) return;
    int p  = idx % P;
    int bt = idx / P;

    int N = 2 * P;
    float px = x[bt * N + 2 * p + 0];                  // points[...,0]
    float py = x[bt * N + 2 * p + 1];                  // points[...,1]
    float res_x = resolution[bt * 2 + 0];
    float res_y = resolution[bt * 2 + 1];
    float org0  = origin[bt * 2 + 0];
    float org1  = origin[bt * 2 + 1];

    // float->int conversion truncates toward zero, matching jnp astype(int32)
    int row = (int)(py / res_y + org0);
    int col = (int)(px / res_x + org1);

    if (row >= 0 && row < H && col >= 0 && col < W) {
        long long flat = ((((long long)bt * H + row) * W) + col) * P + p;
        out[flat] = 1.0f;
    }
}

extern "C" void kernel_launch(void* const* d_in, const int* in_sizes, int n_in,
                              void* d_out, int out_size, void* d_ws, size_t ws_size,
                              hipStream_t stream) {
    const float* x          = (const float*)d_in[0];   // [B,T,N]
    const float* resolution = (const float*)d_in[1];   // [B,T,2]
    const float* origin     = (const float*)d_in[2];   // [B,T,2]
    float* out = (float*)d_out;

    const int B = 8, T = 50, N = 50;
    const int P = N / 2;
    const int H = 100, W = 100;
    const int BT = B * T;

    long long total = (long long)out_size;             // 100,000,000 floats
    long long n4 = total / 4;                          // 25,000,000 float4s
    long long nchunks = n4 / 1024;                     // 24,414 full 16KB chunks
    long long covered = nchunks * 4096;                // floats covered by fill

    int fill_threads = 256;
    int fill_blocks = (int)(nchunks < 16384 ? (nchunks > 0 ? nchunks : 1) : 16384);
    if (nchunks > 0)
        raster_fill_zero<<<fill_blocks, fill_threads, 0, stream>>>((float4v*)out, nchunks);

    if (covered < total) {
        long long rem = total - covered;               // 256 floats here
        int tb = (int)((rem + 255) / 256);
        raster_fill_zero_tail<<<tb, 256, 0, stream>>>(out, covered, total);
    }

    // Scatter: 10,000 lanes total.
    int scat_threads = 256;
    int scat_blocks = (BT * P + scat_threads - 1) / scat_threads;
    raster_scatter<<<scat_blocks, scat_threads, 0, stream>>>(x, resolution, origin, out,
                                                             BT, P, H, W);
}